// AttentionMask_75660143886361
// MI455X (gfx1250) — compile-verified
//
#include <hip/hip_runtime.h>
#include <stdint.h>

// ---------------------------------------------------------------------------
// Sparse coordinate join on MI455X (gfx1250).
// Keys < 2^28  =>  dense bitmap of 1<<28 bits (8M u32 words = 32MB).
// Sorted-unique-union rank of key k == popcount of bits < k  (prefix scan).
// Workspace layout (u32 words):
//   [0            , NWORDS)   union bitmap
//   [NWORDS       , 2*NWORDS) m-bitmap (m keys with score > 0.5)
//   [2*NWORDS     , 3*NWORDS) per-word exclusive popcount prefix
//   [3*NWORDS     , +1024)    block sums
//   [3*NWORDS+1024, +1024)    block exclusive offsets
// Total ~96MB + 8KB.
// ---------------------------------------------------------------------------

#define GRIDC       512
#define NWORDS      (1u << 23)        // 2^28 bits / 32
#define WPB         8192u             // bitmap words per scan block (32KB)
#define SCAN_BLOCKS (NWORDS / WPB)    // 1024
#define THREADS     256

__device__ __forceinline__ uint32_t lds_b32_addr(const void* p) {
  // Flat pointers to LDS carry the workgroup-relative LDS byte offset in the
  // low 32 bits (ISA 10.2 aperture mapping); async-to-LDS VDST wants exactly
  // that.
  return (uint32_t)(uintptr_t)p;
}

// ---------------- zero fill (uint4 stores) ----------------
__global__ void k_zero(uint4* __restrict__ p, size_t n4) {
  size_t i = (size_t)blockIdx.x * blockDim.x + threadIdx.x;
  if (i < n4) p[i] = make_uint4(0u, 0u, 0u, 0u);
}

// ---------------- scatter keys into bitmaps ----------------
__global__ void k_scatter_x(const int* __restrict__ xC,
                            uint32_t* __restrict__ ubm, int Nx) {
  int i = blockIdx.x * blockDim.x + threadIdx.x;
  if (i >= Nx) return;
  int4 c = ((const int4*)xC)[i];
  uint32_t key = (uint32_t)(((c.x * GRIDC + c.y) * GRIDC + c.z) * GRIDC + c.w);
  atomicOr(&ubm[key >> 5], 1u << (key & 31u));
}

__global__ void k_scatter_m(const int* __restrict__ mC,
                            const float* __restrict__ mF,
                            uint32_t* __restrict__ ubm,
                            uint32_t* __restrict__ mbm, int Nm) {
  int i = blockIdx.x * blockDim.x + threadIdx.x;
  if (i >= Nm) return;
  int4 c = ((const int4*)mC)[i];
  uint32_t key = (uint32_t)(((c.x * GRIDC + c.y) * GRIDC + c.z) * GRIDC + c.w);
  uint32_t bit = 1u << (key & 31u);
  atomicOr(&ubm[key >> 5], bit);
  if (mF[i] > 0.5f) atomicOr(&mbm[key >> 5], bit);
}

// ---------------- per-block popcount totals (coalesced) ----------------
__global__ void k_blocksum(const uint32_t* __restrict__ bm,
                           uint32_t* __restrict__ bsum) {
  __shared__ uint32_t red[THREADS];
  size_t base = (size_t)blockIdx.x * WPB;
  uint32_t s = 0;
  for (int k = 0; k < (int)(WPB / THREADS); ++k)
    s += __popc(bm[base + (size_t)k * THREADS + threadIdx.x]);
  red[threadIdx.x] = s;
  __syncthreads();
  for (int off = THREADS / 2; off > 0; off >>= 1) {
    if ((int)threadIdx.x < off) red[threadIdx.x] += red[threadIdx.x + off];
    __syncthreads();
  }
  if (threadIdx.x == 0) bsum[blockIdx.x] = red[0];
}

// ---------------- single-block exclusive scan of 1024 block sums ----------
__global__ void k_scanblocks(const uint32_t* __restrict__ bsum,
                             uint32_t* __restrict__ bexcl) {
  __shared__ uint32_t tmp[SCAN_BLOCKS];
  int t = threadIdx.x;
  uint32_t mine = bsum[t];
  tmp[t] = mine;
  __syncthreads();
  for (int off = 1; off < (int)SCAN_BLOCKS; off <<= 1) {
    uint32_t v = (t >= off) ? tmp[t - off] : 0u;
    __syncthreads();
    tmp[t] += v;
    __syncthreads();
  }
  bexcl[t] = tmp[t] - mine;   // exclusive
}

// ---------------- per-word exclusive prefix, LDS-staged via async copies --
__global__ void k_wordprefix(const uint32_t* __restrict__ bm,
                             const uint32_t* __restrict__ bexcl,
                             uint32_t* __restrict__ wpre) {
  __shared__ uint32_t sw[WPB];
  __shared__ uint32_t ts[THREADS];
  const int t = threadIdx.x;
  const size_t gw = (size_t)blockIdx.x * WPB;

  // Stage this block's 32KB bitmap tile into LDS with gfx1250 async copies:
  // each lane moves 16B per op; 8 iterations x 256 lanes x 16B = 32KB.
  uint64_t gbase = (uint64_t)(uintptr_t)(bm + gw);
  for (int it = 0; it < (int)(WPB / (THREADS * 4)); ++it) {
    uint32_t widx = (uint32_t)(it * THREADS + t) * 4u;       // word index
    uint32_t la = lds_b32_addr(&sw[widx]);
    uint64_t ga = gbase + (uint64_t)widx * 4u;
    asm volatile("global_load_async_to_lds_b128 %0, %1, off"
                 :: "v"(la), "v"(ga) : "memory");
  }
  asm volatile("s_wait_asynccnt 0x0" ::: "memory");
  __syncthreads();

  // Per-thread contiguous chunk of 32 words -> local popcount sum.
  uint32_t s = 0;
  #pragma unroll
  for (int j = 0; j < 32; ++j) s += __popc(sw[t * 32 + j]);
  ts[t] = s;
  __syncthreads();
  // Inclusive Hillis-Steele over 256 thread sums.
  for (int off = 1; off < THREADS; off <<= 1) {
    uint32_t v = (t >= off) ? ts[t - off] : 0u;
    __syncthreads();
    ts[t] += v;
    __syncthreads();
  }
  uint32_t run = bexcl[blockIdx.x] + (ts[t] - s);  // global exclusive base
  #pragma unroll
  for (int j = 0; j < 32; ++j) {
    wpre[gw + (size_t)t * 32 + j] = run;
    run += __popc(sw[t * 32 + j]);
  }
}

// ---------------- emit matched rows ----------------
__global__ void k_emit(const int* __restrict__ xC,
                       const float* __restrict__ xF,
                       const uint32_t* __restrict__ ubm,
                       const uint32_t* __restrict__ mbm,
                       const uint32_t* __restrict__ wpre,
                       float* __restrict__ out, int Nx) {
  int i = blockIdx.x * blockDim.x + threadIdx.x;
  if (i >= Nx) return;
  const float4* src = (const float4*)(xF + (size_t)i * 16);
  __builtin_prefetch(src, 0, 3);   // global_prefetch_b8
  int4 c = ((const int4*)xC)[i];
  uint32_t key = (uint32_t)(((c.x * GRIDC + c.y) * GRIDC + c.z) * GRIDC + c.w);
  uint32_t w = key >> 5, b = key & 31u;
  if (!((mbm[w] >> b) & 1u)) return;             // no m-key with score>0.5
  float4 a0 = src[0], a1 = src[1], a2 = src[2], a3 = src[3];
  bool any = (a0.x > 0.f) | (a0.y > 0.f) | (a0.z > 0.f) | (a0.w > 0.f) |
             (a1.x > 0.f) | (a1.y > 0.f) | (a1.z > 0.f) | (a1.w > 0.f) |
             (a2.x > 0.f) | (a2.y > 0.f) | (a2.z > 0.f) | (a2.w > 0.f) |
             (a3.x > 0.f) | (a3.y > 0.f) | (a3.z > 0.f) | (a3.w > 0.f);
  if (!any) return;
  uint32_t rank = wpre[w] + __popc(ubm[w] & ((1u << b) - 1u));
  float4* dst = (float4*)(out + (size_t)rank * 16);
  dst[0] = a0; dst[1] = a1; dst[2] = a2; dst[3] = a3;
}

// ---------------------------------------------------------------------------
extern "C" void kernel_launch(void* const* d_in, const int* in_sizes, int n_in,
                              void* d_out, int out_size, void* d_ws, size_t ws_size,
                              hipStream_t stream) {
  const int*   xC = (const int*)d_in[0];
  const float* xF = (const float*)d_in[1];
  const int*   mC = (const int*)d_in[2];
  const float* mF = (const float*)d_in[3];
  float* out = (float*)d_out;

  const int Nx = in_sizes[0] / 4;
  const int Nm = in_sizes[2] / 4;

  uint32_t* ubm   = (uint32_t*)d_ws;
  uint32_t* mbm   = ubm + NWORDS;
  uint32_t* wpre  = mbm + NWORDS;
  uint32_t* bsum  = wpre + NWORDS;
  uint32_t* bexcl = bsum + SCAN_BLOCKS;

  // 1) zero both bitmaps (2*NWORDS words) and the output
  {
    size_t n4 = (size_t)(2ull * NWORDS) / 4;
    k_zero<<<(unsigned)((n4 + THREADS - 1) / THREADS), THREADS, 0, stream>>>(
        (uint4*)ubm, n4);
  }
  {
    size_t n4 = (size_t)out_size / 4;
    k_zero<<<(unsigned)((n4 + THREADS - 1) / THREADS), THREADS, 0, stream>>>(
        (uint4*)out, n4);
  }

  // 2) scatter keys into bitmaps
  k_scatter_x<<<(Nx + THREADS - 1) / THREADS, THREADS, 0, stream>>>(xC, ubm, Nx);
  k_scatter_m<<<(Nm + THREADS - 1) / THREADS, THREADS, 0, stream>>>(mC, mF, ubm, mbm, Nm);

  // 3) hierarchical exclusive popcount scan over the union bitmap
  k_blocksum  <<<SCAN_BLOCKS, THREADS, 0, stream>>>(ubm, bsum);
  k_scanblocks<<<1, SCAN_BLOCKS, 0, stream>>>(bsum, bexcl);
  k_wordprefix<<<SCAN_BLOCKS, THREADS, 0, stream>>>(ubm, bexcl, wpre);

  // 4) emit matched feature rows at their union ranks
  k_emit<<<(Nx + THREADS - 1) / THREADS, THREADS, 0, stream>>>(
      xC, xF, ubm, mbm, wpre, out, Nx);
}